// GNNModel_19464791785787
// MI455X (gfx1250) — compile-verified
//
#include <hip/hip_runtime.h>

typedef __attribute__((ext_vector_type(2))) float v2f;
typedef __attribute__((ext_vector_type(8))) float v8f;

#define GNN_F 128
#define GNN_H 256
#define GNN_C 40

#define G1_ROWS 80          // 500000 = 80 * 6250
#define G1_MT 5
#define LDS_X_STRIDE 132    // 128 + 4 pad -> stride%64banks = 4

#define G2_ROWS 80
#define G2_MT 5
#define LDS_A_STRIDE 260    // 256 + 4 pad

// ---------------- unique/rank pipeline ----------------

__global__ void k_init_firstpos(int* __restrict__ first_pos, int n) {
  int i = blockIdx.x * blockDim.x + threadIdx.x;
  if (i < n) first_pos[i] = 0x7FFFFFFF;
}

__global__ void k_firstpos(const int* __restrict__ edge, int* __restrict__ first_pos, int e) {
  int i = blockIdx.x * blockDim.x + threadIdx.x;
  if (i < e) atomicMin(&first_pos[edge[2 * i]], i);
}

// flag[p] = (first_pos[seg[p]] == p); per-block exclusive scan of flags
__global__ __launch_bounds__(512) void k_flag_scan(const int* __restrict__ edge,
                                                   const int* __restrict__ first_pos,
                                                   int* __restrict__ scanbuf,
                                                   int* __restrict__ blocksums, int e) {
  __shared__ int s[512];
  const int tid = threadIdx.x;
  const int p = blockIdx.x * 512 + tid;
  int f = 0;
  if (p < e) {
    int v = edge[2 * p];
    f = (first_pos[v] == p) ? 1 : 0;
  }
  s[tid] = f;
  __syncthreads();
  for (int d = 1; d < 512; d <<= 1) {
    int t = (tid >= d) ? s[tid - d] : 0;
    __syncthreads();
    s[tid] += t;
    __syncthreads();
  }
  if (p < e) scanbuf[p] = s[tid] - f;  // exclusive
  if (tid == 511) blocksums[blockIdx.x] = s[511];
}

__global__ __launch_bounds__(1024) void k_scan_sums(int* __restrict__ blocksums, int nb) {
  __shared__ int s[1024];
  const int tid = threadIdx.x;
  int v = (tid < nb) ? blocksums[tid] : 0;
  s[tid] = v;
  __syncthreads();
  for (int d = 1; d < 1024; d <<= 1) {
    int t = (tid >= d) ? s[tid - d] : 0;
    __syncthreads();
    s[tid] += t;
    __syncthreads();
  }
  if (tid < nb) blocksums[tid] = s[tid] - v;  // exclusive
}

__global__ void k_scan_add(int* __restrict__ scanbuf, const int* __restrict__ blocksums, int e) {
  int p = blockIdx.x * blockDim.x + threadIdx.x;
  if (p < e) scanbuf[p] += blocksums[p / 512];
}

// mapped[i] = rank[seg[seg[i]]] = scan[first_pos[edge2[edge2[i]]]]
__global__ void k_map(const int* __restrict__ edge, const int* __restrict__ first_pos,
                      const int* __restrict__ scanbuf, int* __restrict__ mapped, int e) {
  int i = blockIdx.x * blockDim.x + threadIdx.x;
  if (i < e) {
    int s1 = edge[2 * i];
    int s2 = edge[2 * s1];
    mapped[i] = scanbuf[first_pos[s2]];
  }
}

__global__ void k_zero4(float4* __restrict__ p, size_t n4) {
  size_t i = (size_t)blockIdx.x * blockDim.x + threadIdx.x;
  if (i < n4) p[i] = make_float4(0.f, 0.f, 0.f, 0.f);
}

// Async global->LDS copy of 16 bytes (per lane), ASYNCcnt-tracked.
__device__ __forceinline__ void async_copy_b128(void* lds_dst, const void* gsrc) {
  unsigned lds_off = (unsigned)(uintptr_t)lds_dst;  // LDS aperture: addr[31:0] = LDS offset
  asm volatile("global_load_async_to_lds_b128 %0, %1, off"
               :: "v"(lds_off), "v"(gsrc) : "memory");
}
__device__ __forceinline__ void async_copy_wait() {
  asm volatile("s_wait_asynccnt 0x0" ::: "memory");
}

// ---------------- GEMM1 (x@W1 + b1, relu) fused with segment scatter-add ----------------
// 1 block = 80 rows (5 m-tiles), 8 waves; wave w owns cols [32w, 32w+32) (2 n-tiles).
// Each B fragment is reused across 5 m-tiles -> 10 WMMA per k-step per wave.
__global__ __launch_bounds__(256) void k_gemm1_scatter(
    const float* __restrict__ x, const float* __restrict__ W1,
    const float* __restrict__ b1, const int* __restrict__ mapped,
    float* __restrict__ agg) {
  __shared__ float sX[G1_ROWS * LDS_X_STRIDE];  // ~41.25 KB, padded stride
  const int tid = threadIdx.x;
  const int row0 = blockIdx.x * G1_ROWS;

  // stage 80x128 x-tile via async global->LDS (10 x b128 per thread)
  for (int c = tid; c < G1_ROWS * (GNN_F / 4); c += 256) {
    const int r = c >> 5, c4 = c & 31;
    async_copy_b128(&sX[r * LDS_X_STRIDE + c4 * 4],
                    x + (size_t)(row0 + r) * GNN_F + c4 * 4);
  }
  async_copy_wait();
  __syncthreads();

  const int wave = tid >> 5;
  const int lane = tid & 31;
  const int lh = lane >> 4;  // half-wave selects K pair {0,1} vs {2,3}
  const int ll = lane & 15;  // row (A) / col (B,C)
  const int n0 = wave * 32;

  v8f acc[G1_MT][2];
#pragma unroll
  for (int mt = 0; mt < G1_MT; ++mt) {
    acc[mt][0] = (v8f){};
    acc[mt][1] = (v8f){};
  }

  for (int k = 0; k < GNN_F; k += 4) {
    const int ka = k + 2 * lh;
    v2f bA, bB;
    bA[0] = W1[(size_t)ka * GNN_H + n0 + ll];
    bA[1] = W1[(size_t)(ka + 1) * GNN_H + n0 + ll];
    bB[0] = W1[(size_t)ka * GNN_H + n0 + 16 + ll];
    bB[1] = W1[(size_t)(ka + 1) * GNN_H + n0 + 16 + ll];
#pragma unroll
    for (int mt = 0; mt < G1_MT; ++mt) {
      v2f a;
      a[0] = sX[(mt * 16 + ll) * LDS_X_STRIDE + ka];
      a[1] = sX[(mt * 16 + ll) * LDS_X_STRIDE + ka + 1];
      acc[mt][0] = __builtin_amdgcn_wmma_f32_16x16x4_f32(false, a, false, bA, (short)0,
                                                         acc[mt][0], false, false);
      acc[mt][1] = __builtin_amdgcn_wmma_f32_16x16x4_f32(false, a, false, bB, (short)0,
                                                         acc[mt][1], false, false);
    }
  }

  const float bias0 = b1[n0 + ll];
  const float bias1 = b1[n0 + 16 + ll];
#pragma unroll
  for (int mt = 0; mt < G1_MT; ++mt) {
#pragma unroll
    for (int r = 0; r < 8; ++r) {
      const int grow = row0 + mt * 16 + r + 8 * lh;  // C/D layout: VGPR r -> rows r, r+8
      const int dest = mapped[grow];
      float v0 = fmaxf(acc[mt][0][r] + bias0, 0.f);
      float v1 = fmaxf(acc[mt][1][r] + bias1, 0.f);
      atomicAdd(&agg[(size_t)dest * GNN_H + n0 + ll], v0);
      atomicAdd(&agg[(size_t)dest * GNN_H + n0 + 16 + ll], v1);
    }
  }
}

// ---------------- GEMM2: out = relu(agg) @ W2 + b2 ----------------
// 1 block = 80 rows (5 m-tiles), 3 waves = 3 n-tiles (cols 0..47, guard n<40).
// relu applied when reading A fragments from LDS (co-executes with WMMA).
__global__ __launch_bounds__(96) void k_gemm2(
    const float* __restrict__ agg, const float* __restrict__ W2,
    const float* __restrict__ b2, float* __restrict__ out) {
  __shared__ float sA[G2_ROWS * LDS_A_STRIDE];  // ~81.25 KB, padded stride
  const int tid = threadIdx.x;
  const int row0 = blockIdx.x * G2_ROWS;

  // stage 80x256 agg-tile via async global->LDS (<=54 x b128 per thread)
  for (int c = tid; c < G2_ROWS * (GNN_H / 4); c += 96) {
    const int r = c >> 6, c4 = c & 63;
    async_copy_b128(&sA[r * LDS_A_STRIDE + c4 * 4],
                    agg + (size_t)(row0 + r) * GNN_H + c4 * 4);
  }
  async_copy_wait();
  __syncthreads();

  const int wave = tid >> 5;
  const int lane = tid & 31;
  const int lh = lane >> 4;
  const int ll = lane & 15;
  const int n = wave * 16 + ll;
  const bool nok = (n < GNN_C);

  v8f acc[G2_MT];
#pragma unroll
  for (int mt = 0; mt < G2_MT; ++mt) acc[mt] = (v8f){};

  for (int k = 0; k < GNN_H; k += 4) {
    const int ka = k + 2 * lh;
    v2f b;
    b[0] = nok ? W2[(size_t)ka * GNN_C + n] : 0.f;
    b[1] = nok ? W2[(size_t)(ka + 1) * GNN_C + n] : 0.f;
#pragma unroll
    for (int mt = 0; mt < G2_MT; ++mt) {
      v2f a;
      a[0] = fmaxf(sA[(mt * 16 + ll) * LDS_A_STRIDE + ka], 0.f);
      a[1] = fmaxf(sA[(mt * 16 + ll) * LDS_A_STRIDE + ka + 1], 0.f);
      acc[mt] = __builtin_amdgcn_wmma_f32_16x16x4_f32(false, a, false, b, (short)0,
                                                      acc[mt], false, false);
    }
  }

  if (nok) {
    const float bias = b2[n];
#pragma unroll
    for (int mt = 0; mt < G2_MT; ++mt) {
#pragma unroll
      for (int r = 0; r < 8; ++r) {
        const int grow = row0 + mt * 16 + r + 8 * lh;
        out[(size_t)grow * GNN_C + n] = acc[mt][r] + bias;
      }
    }
  }
}

// ---------------- host-side launch ----------------

extern "C" void kernel_launch(void* const* d_in, const int* in_sizes, int n_in,
                              void* d_out, int out_size, void* d_ws, size_t ws_size,
                              hipStream_t stream) {
  const float* x  = (const float*)d_in[0];
  const int* edge = (const int*)d_in[1];
  const float* W1 = (const float*)d_in[2];
  const float* b1 = (const float*)d_in[3];
  const float* W2 = (const float*)d_in[4];
  const float* b2 = (const float*)d_in[5];
  float* out = (float*)d_out;

  const int N = in_sizes[0] / GNN_F;  // 500000
  const int E = in_sizes[1] / 2;      // 500000

  char* ws = (char*)d_ws;
  size_t off = 0;
  int* first_pos = (int*)(ws + off); off += (size_t)N * sizeof(int);
  int* scanbuf   = (int*)(ws + off); off += (size_t)E * sizeof(int);
  int* blocksums = (int*)(ws + off); off += 4096;
  int* mapped    = (int*)(ws + off); off += (size_t)E * sizeof(int);
  float* agg     = (float*)(ws + off); off += (size_t)N * GNN_H * sizeof(float);

  const int nScanBlocks = (E + 511) / 512;  // 977 <= 1024

  k_init_firstpos<<<(N + 255) / 256, 256, 0, stream>>>(first_pos, N);
  k_firstpos<<<(E + 255) / 256, 256, 0, stream>>>(edge, first_pos, E);
  k_flag_scan<<<nScanBlocks, 512, 0, stream>>>(edge, first_pos, scanbuf, blocksums, E);
  k_scan_sums<<<1, 1024, 0, stream>>>(blocksums, nScanBlocks);
  k_scan_add<<<(E + 255) / 256, 256, 0, stream>>>(scanbuf, blocksums, E);
  k_map<<<(E + 255) / 256, 256, 0, stream>>>(edge, first_pos, scanbuf, mapped, E);

  const size_t n4 = (size_t)N * GNN_H / 4;
  k_zero4<<<(unsigned)((n4 + 255) / 256), 256, 0, stream>>>((float4*)agg, n4);

  k_gemm1_scatter<<<N / G1_ROWS, 256, 0, stream>>>(x, W1, b1, mapped, agg);
  k_gemm2<<<N / G2_ROWS, 96, 0, stream>>>(agg, W2, b2, out);
}